// DistributedMemory_46325517254816
// MI455X (gfx1250) — compile-verified
//
#include <hip/hip_runtime.h>

#define VEC_DIM 128
#define N_WORDS 100000
#define CTX 8
#define NSAMP 10
#define BATCH 16384

#define WAVES_PER_BLOCK 4
#define ROW_STRIDE 132   // 128 + 4 pad: 16B-aligned rows, bank-conflict-free A reads

typedef __attribute__((ext_vector_type(2))) float v2f;
typedef __attribute__((ext_vector_type(4))) float v4f;
typedef __attribute__((ext_vector_type(8))) float v8f;

// One wave32 handles a group of 16 batch rows end-to-end:
//  Phase 1: build 16x128 "inputs" tile in LDS (coalesced float4 gathers)
//  Phase 2: pull A into VGPRs in V_WMMA_F32_16X16X4_F32 A-layout
//  Phase 3: per sample slot, gather B columns and chain 32 WMMAs over K=128,
//           extract the diagonal of the 16x16 product via LDS.
__global__ __launch_bounds__(WAVES_PER_BLOCK * 32) void pv_dm_wmma_kernel(
    const int*   __restrict__ doc_ids,      // [BATCH]
    const int*   __restrict__ context_ids,  // [BATCH, CTX]
    const int*   __restrict__ sample_ids,   // [BATCH, NSAMP]
    const float* __restrict__ paragraph,    // [N_DOCS, 128]
    const float* __restrict__ words,        // [N_WORDS, 128]
    const float* __restrict__ outW,         // [128, N_WORDS]
    float*       __restrict__ logits)       // [BATCH, NSAMP]
{
    __shared__ float sA[WAVES_PER_BLOCK * 16 * ROW_STRIDE];
    __shared__ float sD[WAVES_PER_BLOCK * 256];

    const int lane  = threadIdx.x & 31;
    const int wave  = threadIdx.x >> 5;
    const int group = blockIdx.x * WAVES_PER_BLOCK + wave;  // 0..1023
    const int b0    = group * 16;

    float* A = &sA[wave * 16 * ROW_STRIDE];
    float* D = &sD[wave * 256];

    // ---- Phase 1: inputs tile, fully coalesced (32 lanes x float4 = 128 floats/row)
    const int l4 = lane * 4;
    for (int r = 0; r < 16; ++r) {
        const int b   = b0 + r;
        const int doc = doc_ids[b];
        v4f acc = *(const v4f*)(paragraph + (size_t)doc * VEC_DIM + l4);
#pragma unroll
        for (int c = 0; c < CTX; ++c) {
            const int w = context_ids[b * CTX + c];
            acc += *(const v4f*)(words + (size_t)w * VEC_DIM + l4);
        }
        *(v4f*)(A + r * ROW_STRIDE + l4) = acc;
    }
    asm volatile("s_wait_dscnt 0" ::: "memory");   // wave-local LDS RAW

    // ---- Phase 2: A-matrix VGPR layout for V_WMMA_F32_16X16X4_F32
    // lanes 0-15: row m=lane, K pair {kbase+0, kbase+1}
    // lanes 16-31: row m=lane-16, K pair {kbase+2, kbase+3}
    const int m  = lane & 15;
    const int kb = (lane >> 4) << 1;  // 0 or 2
    v2f areg[32];
#pragma unroll
    for (int j = 0; j < 32; ++j)
        areg[j] = *(const v2f*)(A + m * ROW_STRIDE + 4 * j + kb);
    asm volatile("s_wait_dscnt 0" ::: "memory");

    // ---- Phase 3: per sample slot, B = 16 gathered columns of outW
    for (int s = 0; s < NSAMP; ++s) {
        const int sid = sample_ids[(b0 + m) * NSAMP + s];   // column for N-index m
        const float* col = outW + sid;

        v8f c = {0.f, 0.f, 0.f, 0.f, 0.f, 0.f, 0.f, 0.f};
#pragma unroll
        for (int j = 0; j < 32; ++j) {
            const int k = 4 * j + kb;
            // B-matrix layout (4x16 f32): VGPR0 holds K={kb}, VGPR1 holds K={kb+1}
            v2f bv;
            bv.x = col[(size_t)k       * N_WORDS];
            bv.y = col[(size_t)(k + 1) * N_WORDS];
            c = __builtin_amdgcn_wmma_f32_16x16x4_f32(
                    /*neg_a=*/false, areg[j], /*neg_b=*/false, bv,
                    /*c_mod=*/(short)0, c, /*reuse_a=*/false, /*reuse_b=*/false);
        }

        // Diagonal extraction via LDS (transposed store: conflict-free)
#pragma unroll
        for (int t = 0; t < 8; ++t)
            D[t * 32 + lane] = c[t];
        asm volatile("s_wait_dscnt 0" ::: "memory");

        if (lane < 16) {
            // element (M=N=lane): M<8 -> D[33*M]; M>=8 -> D[33*M-240]
            const float dval = (lane < 8) ? D[33 * lane] : D[33 * lane - 240];
            logits[(b0 + lane) * NSAMP + s] = dval;
        }
        asm volatile("s_wait_dscnt 0" ::: "memory");
    }
}

extern "C" void kernel_launch(void* const* d_in, const int* in_sizes, int n_in,
                              void* d_out, int out_size, void* d_ws, size_t ws_size,
                              hipStream_t stream) {
    const int*   doc_ids     = (const int*)d_in[0];
    const int*   context_ids = (const int*)d_in[1];
    const int*   sample_ids  = (const int*)d_in[2];
    const float* paragraph   = (const float*)d_in[3];
    const float* words       = (const float*)d_in[4];
    const float* outW        = (const float*)d_in[5];
    float*       logits      = (float*)d_out;

    const int groups = BATCH / 16;                       // 1024
    dim3 grid(groups / WAVES_PER_BLOCK);                 // 256 blocks
    dim3 block(WAVES_PER_BLOCK * 32);                    // 128 threads = 4 waves

    hipLaunchKernelGGL(pv_dm_wmma_kernel, grid, block, 0, stream,
                       doc_ids, context_ids, sample_ids, paragraph, words, outW,
                       logits);
}